// GATEncoder_25039659336309
// MI455X (gfx1250) — compile-verified
//
#include <hip/hip_runtime.h>
#include <math.h>

// Problem constants (from reference)
#define N_NODES 50000
#define N_EDGES 800000
#define HIDW    128
#define NHEAD   4
#define CPH     32
#define NLAYERS 3
#define EDIMW   3

typedef __attribute__((ext_vector_type(2))) float v2f;
typedef __attribute__((ext_vector_type(8))) float v8f;

// ---------------------------------------------------------------------------
// Order-preserving float<->uint encoding for atomicMax-based segment max
// ---------------------------------------------------------------------------
__device__ __forceinline__ unsigned encOrd(float f) {
  unsigned u = __float_as_uint(f);
  return (u & 0x80000000u) ? ~u : (u | 0x80000000u);
}
__device__ __forceinline__ float decOrd(unsigned u) {
  return __uint_as_float((u & 0x80000000u) ? (u & 0x7FFFFFFFu) : ~u);
}
#define ENC_NEG_INF 0x007FFFFFu  // encOrd(-inf)

// ---------------------------------------------------------------------------
// Per-layer init: zero out-accumulator, denom, stats; lmax = enc(-inf)
// ---------------------------------------------------------------------------
__global__ __launch_bounds__(256)
void init_layer_kernel(float* __restrict__ hout, unsigned* __restrict__ lmax,
                       float* __restrict__ denom, float* __restrict__ stats) {
  int i = blockIdx.x * 256 + threadIdx.x;
  if (i < N_NODES * HIDW) hout[i] = 0.f;
  if (i < N_NODES * NHEAD) { lmax[i] = ENC_NEG_INF; denom[i] = 0.f; }
  if (i < 2) stats[i] = 0.f;
}

// ---------------------------------------------------------------------------
// Dual fp32 WMMA GEMM:  O0 = A@W0, O1 = A@W1   (A: [N,128], W: [128,128])
// One block = 16 output rows; 8 waves each own a 16-col tile; A staged in LDS
// (padded stride 132 to dodge LDS bank conflicts); A fragment reused for two
// v_wmma_f32_16x16x4_f32 accumulations per K step.
// ---------------------------------------------------------------------------
__global__ __launch_bounds__(256)
void gemm_dual_wmma(const float* __restrict__ A,
                    const float* __restrict__ W0,
                    const float* __restrict__ W1,
                    float* __restrict__ O0,
                    float* __restrict__ O1) {
  __shared__ float As[16 * 132];
  const int rowBase = blockIdx.x * 16;
  for (int i = threadIdx.x; i < 16 * HIDW; i += 256) {
    int r = i >> 7, c = i & 127;
    As[r * 132 + c] = A[(rowBase + r) * HIDW + c];
  }
  __syncthreads();
  const int wave    = threadIdx.x >> 5;
  const int lane    = threadIdx.x & 31;
  const int colBase = wave * 16;
  const int laneHi  = lane >> 4;   // selects K pair (0: K0/K1, 1: K2/K3)
  const int laneLo  = lane & 15;   // A: row M ; B/C/D: col N
  v8f c0 = {}; v8f c1 = {};
  for (int k0 = 0; k0 < HIDW; k0 += 4) {
    const int ka = k0 + 2 * laneHi;
    v2f a;
    a.x = As[laneLo * 132 + ka];
    a.y = As[laneLo * 132 + ka + 1];
    v2f b0, b1;
    b0.x = W0[(ka + 0) * HIDW + colBase + laneLo];
    b0.y = W0[(ka + 1) * HIDW + colBase + laneLo];
    b1.x = W1[(ka + 0) * HIDW + colBase + laneLo];
    b1.y = W1[(ka + 1) * HIDW + colBase + laneLo];
    c0 = __builtin_amdgcn_wmma_f32_16x16x4_f32(false, a, false, b0, (short)0, c0, false, false);
    c1 = __builtin_amdgcn_wmma_f32_16x16x4_f32(false, a, false, b1, (short)0, c1, false, false);
  }
#pragma unroll
  for (int r = 0; r < 8; ++r) {
    const int row = rowBase + r + 8 * laneHi;
    O0[row * HIDW + colBase + laneLo] = c0[r];
    O1[row * HIDW + colBase + laneLo] = c1[r];
  }
}

// Single WMMA GEMM with bias for the final projection: O = A@W + b
__global__ __launch_bounds__(256)
void gemm_bias_wmma(const float* __restrict__ A,
                    const float* __restrict__ W,
                    const float* __restrict__ bias,
                    float* __restrict__ O) {
  __shared__ float As[16 * 132];
  const int rowBase = blockIdx.x * 16;
  for (int i = threadIdx.x; i < 16 * HIDW; i += 256) {
    int r = i >> 7, c = i & 127;
    As[r * 132 + c] = A[(rowBase + r) * HIDW + c];
  }
  __syncthreads();
  const int wave    = threadIdx.x >> 5;
  const int lane    = threadIdx.x & 31;
  const int colBase = wave * 16;
  const int laneHi  = lane >> 4;
  const int laneLo  = lane & 15;
  v8f c0 = {};
  for (int k0 = 0; k0 < HIDW; k0 += 4) {
    const int ka = k0 + 2 * laneHi;
    v2f a;
    a.x = As[laneLo * 132 + ka];
    a.y = As[laneLo * 132 + ka + 1];
    v2f b;
    b.x = W[(ka + 0) * HIDW + colBase + laneLo];
    b.y = W[(ka + 1) * HIDW + colBase + laneLo];
    c0 = __builtin_amdgcn_wmma_f32_16x16x4_f32(false, a, false, b, (short)0, c0, false, false);
  }
  const float bv = bias[colBase + laneLo];
#pragma unroll
  for (int r = 0; r < 8; ++r) {
    const int row = rowBase + r + 8 * laneHi;
    O[row * HIDW + colBase + laneLo] = c0[r] + bv;
  }
}

// ---------------------------------------------------------------------------
// Pass A: per-edge GATv2 logits + segment max (one wave32 per edge;
// lane = channel-in-head, 4 heads unrolled; butterfly reduce for dots).
// Fuses the edge_attr@We transform (K=3) from LDS-resident weights.
// ---------------------------------------------------------------------------
__global__ __launch_bounds__(256)
void edge_logits_kernel(const int* __restrict__ ei, const float* __restrict__ eattr,
                        const float* __restrict__ xl, const float* __restrict__ xr,
                        const float* __restrict__ We, const float* __restrict__ att,
                        float* __restrict__ elog, unsigned* __restrict__ lmaxu) {
  __shared__ float sWe[EDIMW * HIDW];
  __shared__ float sAtt[HIDW];
  for (int i = threadIdx.x; i < EDIMW * HIDW; i += 256) sWe[i] = We[i];
  if (threadIdx.x < HIDW) sAtt[threadIdx.x] = att[threadIdx.x];
  __syncthreads();
  const int lane = threadIdx.x & 31;
  const int wid  = (blockIdx.x * 256 + threadIdx.x) >> 5;
  const int nw   = (gridDim.x * 256) >> 5;
  for (int e = wid; e < N_EDGES; e += nw) {
    const int src = ei[e];
    const int dst = ei[N_EDGES + e];
    float eav = (lane < 3) ? eattr[e * 3 + lane] : 0.f;
    const float ea0 = __shfl(eav, 0, 32);
    const float ea1 = __shfl(eav, 1, 32);
    const float ea2 = __shfl(eav, 2, 32);
    float acc[NHEAD];
#pragma unroll
    for (int h = 0; h < NHEAD; ++h) {
      const int c = h * CPH + lane;
      float eac = ea0 * sWe[c] + ea1 * sWe[HIDW + c] + ea2 * sWe[2 * HIDW + c];
      float v = xl[src * HIDW + c] + xr[dst * HIDW + c] + eac;
      v = (v > 0.f) ? v : 0.2f * v;  // leaky_relu
      acc[h] = v * sAtt[c];
    }
#pragma unroll
    for (int off = 16; off > 0; off >>= 1) {
#pragma unroll
      for (int h = 0; h < NHEAD; ++h) acc[h] += __shfl_xor(acc[h], off, 32);
    }
    if (lane < NHEAD) {
      float lg = (lane == 0) ? acc[0] : (lane == 1) ? acc[1] : (lane == 2) ? acc[2] : acc[3];
      elog[e * NHEAD + lane] = lg;
      atomicMax(&lmaxu[dst * NHEAD + lane], encOrd(lg));
    }
  }
}

// Decode encoded max; non-finite (no incoming edges) -> 0 (matches reference)
__global__ __launch_bounds__(256)
void lmax_fixup_kernel(unsigned* __restrict__ lmaxu) {
  int i = blockIdx.x * 256 + threadIdx.x;
  if (i >= N_NODES * NHEAD) return;
  float f = decOrd(lmaxu[i]);
  ((float*)lmaxu)[i] = isfinite(f) ? f : 0.f;
}

// Pass B: ex = exp(logit - lmax[dst]); accumulate softmax denominator
__global__ __launch_bounds__(256)
void edge_expsum_kernel(const int* __restrict__ ei, float* __restrict__ elog,
                        const float* __restrict__ lmaxf, float* __restrict__ denom) {
  int t = blockIdx.x * 256 + threadIdx.x;
  if (t >= N_EDGES * NHEAD) return;
  int e = t >> 2, h = t & 3;
  int dst = ei[N_EDGES + e];
  float ex = expf(elog[t] - lmaxf[dst * NHEAD + h]);
  elog[t] = ex;
  atomicAdd(&denom[dst * NHEAD + h], ex);
}

// Pass C: msg = xl[src] * alpha, scatter-add into out[dst] (L2-resident)
__global__ __launch_bounds__(256)
void edge_scatter_kernel(const int* __restrict__ ei, const float* __restrict__ xl,
                         const float* __restrict__ ex, const float* __restrict__ denom,
                         float* __restrict__ out) {
  const int lane = threadIdx.x & 31;
  const int wid  = (blockIdx.x * 256 + threadIdx.x) >> 5;
  const int nw   = (gridDim.x * 256) >> 5;
  for (int e = wid; e < N_EDGES; e += nw) {
    const int src = ei[e];
    const int dst = ei[N_EDGES + e];
    float exv = 0.f, dnv = 0.f;
    if (lane < NHEAD) {
      exv = ex[e * NHEAD + lane];
      dnv = denom[dst * NHEAD + lane];
    }
#pragma unroll
    for (int h = 0; h < NHEAD; ++h) {
      const float alpha = __shfl(exv, h, 32) / (__shfl(dnv, h, 32) + 1e-16f);
      const int c = h * CPH + lane;
      atomicAdd(&out[dst * HIDW + c], xl[src * HIDW + c] * alpha);
    }
  }
}

// Graph LayerNorm stats: global sum & sumsq over all N*HID elements
__global__ __launch_bounds__(256)
void ln_reduce_kernel(const float* __restrict__ h, float* __restrict__ stats) {
  __shared__ float s1[256], s2[256];
  float a = 0.f, b = 0.f;
  for (int i = blockIdx.x * 256 + threadIdx.x; i < N_NODES * HIDW; i += gridDim.x * 256) {
    float v = h[i];
    a += v; b += v * v;
  }
  s1[threadIdx.x] = a; s2[threadIdx.x] = b;
  __syncthreads();
  for (int off = 128; off > 0; off >>= 1) {
    if ((int)threadIdx.x < off) {
      s1[threadIdx.x] += s1[threadIdx.x + off];
      s2[threadIdx.x] += s2[threadIdx.x + off];
    }
    __syncthreads();
  }
  if (threadIdx.x == 0) { atomicAdd(&stats[0], s1[0]); atomicAdd(&stats[1], s2[0]); }
}

// Apply graph LayerNorm + exact GELU in place
__global__ __launch_bounds__(256)
void ln_gelu_kernel(float* __restrict__ h, const float* __restrict__ stats,
                    const float* __restrict__ w, const float* __restrict__ b) {
  int i = blockIdx.x * 256 + threadIdx.x;
  if (i >= N_NODES * HIDW) return;
  const float inv  = 1.0f / (float)(N_NODES * HIDW);
  const float mean = stats[0] * inv;
  const float var  = stats[1] * inv - mean * mean;
  const float rs   = rsqrtf(var + 1e-5f);
  const int c = i & (HIDW - 1);
  float v = (h[i] - mean) * rs * w[c] + b[c];
  v = 0.5f * v * (1.0f + erff(v * 0.70710678118654752f));
  h[i] = v;
}

// ---------------------------------------------------------------------------
extern "C" void kernel_launch(void* const* d_in, const int* in_sizes, int n_in,
                              void* d_out, int out_size, void* d_ws, size_t ws_size,
                              hipStream_t stream) {
  const float* x     = (const float*)d_in[0];
  const int*   ei    = (const int*)  d_in[1];
  const float* eattr = (const float*)d_in[2];
  const float* Wl    = (const float*)d_in[3];
  const float* Wr    = (const float*)d_in[4];
  const float* We    = (const float*)d_in[5];
  const float* att   = (const float*)d_in[6];
  const float* lnw   = (const float*)d_in[7];
  const float* lnb   = (const float*)d_in[8];
  const float* Wout  = (const float*)d_in[9];
  const float* bout  = (const float*)d_in[10];
  float* out = (float*)d_out;

  // Carve scratch (~117 MB): hA/hB ping-pong, xl, xr, edge buf, seg buffers
  char* p = (char*)d_ws;
  auto carve = [&](size_t bytes) -> void* {
    void* q = (void*)p;
    p += (bytes + 255) & ~(size_t)255;
    return q;
  };
  float*    hA    = (float*)   carve((size_t)N_NODES * HIDW * 4);
  float*    hB    = (float*)   carve((size_t)N_NODES * HIDW * 4);
  float*    xl    = (float*)   carve((size_t)N_NODES * HIDW * 4);
  float*    xr    = (float*)   carve((size_t)N_NODES * HIDW * 4);
  float*    elog  = (float*)   carve((size_t)N_EDGES * NHEAD * 4);
  unsigned* lmax  = (unsigned*)carve((size_t)N_NODES * NHEAD * 4);
  float*    denom = (float*)   carve((size_t)N_NODES * NHEAD * 4);
  float*    stats = (float*)   carve(2 * 4);

  const int elemBlocks = (N_NODES * HIDW + 255) / 256;       // 25000
  const int segBlocks  = (N_NODES * NHEAD + 255) / 256;
  const int ehBlocks   = (N_EDGES * NHEAD + 255) / 256;      // 12500
  const int waveBlocks = 25000;                              // 200k waves, ~4 edges each
  const int gemmBlocks = N_NODES / 16;                       // 3125 (exact)

  const float* hcur = x;
  float* bufs[2] = { hA, hB };
  for (int l = 0; l < NLAYERS; ++l) {
    float* hout = bufs[l & 1];
    init_layer_kernel<<<elemBlocks, 256, 0, stream>>>(hout, lmax, denom, stats);
    gemm_dual_wmma<<<gemmBlocks, 256, 0, stream>>>(hcur, Wl + (size_t)l * HIDW * HIDW,
                                                   Wr + (size_t)l * HIDW * HIDW, xl, xr);
    edge_logits_kernel<<<waveBlocks, 256, 0, stream>>>(ei, eattr, xl, xr,
                                                       We + (size_t)l * EDIMW * HIDW,
                                                       att + (size_t)l * HIDW, elog, lmax);
    lmax_fixup_kernel<<<segBlocks, 256, 0, stream>>>(lmax);
    edge_expsum_kernel<<<ehBlocks, 256, 0, stream>>>(ei, elog, (const float*)lmax, denom);
    edge_scatter_kernel<<<waveBlocks, 256, 0, stream>>>(ei, xl, elog, denom, hout);
    ln_reduce_kernel<<<2048, 256, 0, stream>>>(hout, stats);
    ln_gelu_kernel<<<elemBlocks, 256, 0, stream>>>(hout, stats,
                                                   lnw + (size_t)l * HIDW,
                                                   lnb + (size_t)l * HIDW);
    hcur = hout;
  }
  gemm_bias_wmma<<<gemmBlocks, 256, 0, stream>>>(hcur, Wout, bout, out);
}